// JosephsonJunctionSDE_31275951849821
// MI455X (gfx1250) — compile-verified
//
#include <hip/hip_runtime.h>

// Euler-Maruyama for the coupled Josephson junction SDE on gfx1250.
// One thread per trajectory; noise double-buffered into LDS via the CDNA5
// async global->LDS data mover (GLOBAL_LOAD_ASYNC_TO_LDS_B64 / ASYNCcnt).
//
// Structure per chunk (32 steps):
//   s_wait_asynccnt 0          <- chunk c's DMA complete (only thing in flight)
//   issue chunk c+1 DMA burst  <- 32x global_load_async_to_lds_b64, unrolled
//   compute 32 steps           <- ds_load_b64 noise, sin/FMA, global_store_b128
// so the next chunk's HBM traffic streams while this chunk's dependent
// integration chain runs on the VALU.

#define TCHUNK 32   // time steps per chunk (32 async b64 loads in flight)
#define TPB    128  // threads per block (4 wave32s)

// 32-bit LDS byte offset of a generic pointer to a __shared__ object
// (VDST operand of global_load_async_to_lds_* carries the LDS byte address).
__device__ __forceinline__ unsigned lds_off_u32(const void* p) {
  return (unsigned)(unsigned long long)
      (__attribute__((address_space(3))) const void*)p;
}

__global__ __launch_bounds__(TPB) void jj_sde_kernel(
    const float* __restrict__ params,
    const float* __restrict__ y0,
    const float* __restrict__ noise,   // [(N-1), B, 2] f32
    const int*   __restrict__ Tp,      // scalar T
    const int*   __restrict__ Np,      // scalar N
    float*       __restrict__ out,     // [B, N, 4] f32
    int batchTotal)
{
  // Double-buffered noise staging: [buf][step][thread][2] floats = 64 KB.
  __shared__ __align__(16) float lds_noise[2][TCHUNK][TPB][2];

  const int tid = threadIdx.x;
  const int b   = blockIdx.x * TPB + tid;
  const bool active = (b < batchTotal);
  const int bc = active ? b : 0;  // clamp so inactive lanes issue safe addresses

  const int   N      = Np[0];
  const int   nsteps = N - 1;
  const float dt      = (float)Tp[0] / (float)nsteps;
  const float sqrt_dt = __builtin_sqrtf(dt);

  const float beta1 = params[0], beta2 = params[1];
  const float i1    = params[2], i2    = params[3];
  const float kap1  = params[4], kap2  = params[5];
  const float s1    = params[6] * sqrt_dt;   // sigma1 * sqrt(dt)
  const float s2    = params[7] * sqrt_dt;   // sigma2 * sqrt(dt)

  // Load initial state and emit trajectory sample n = 0.
  float phi1 = 0.f, v1 = 0.f, phi2 = 0.f, v2 = 0.f;
  if (active) {
    const float4 y = *reinterpret_cast<const float4*>(y0 + (size_t)b * 4);
    phi1 = y.x; v1 = y.y; phi2 = y.z; v2 = y.w;
    *reinterpret_cast<float4*>(out + (size_t)b * N * 4) = y;
  }

  const int full    = nsteps / TCHUNK;          // chunks with exactly TCHUNK steps
  const int tail    = nsteps - full * TCHUNK;   // 0..TCHUNK-1 leftover steps
  const int nchunks = full + (tail ? 1 : 0);

  const unsigned gstride = (unsigned)batchTotal * 8u;  // bytes per step row

  // Issue one async global->LDS b64 load: noise[k][b][0..1] -> LDS slot.
  auto issue_one = [&](int buf, int kl, unsigned goff) {
    const unsigned loff = lds_off_u32(&lds_noise[buf][kl][tid][0]);
    asm volatile("global_load_async_to_lds_b64 %0, %1, %2"
                 :: "v"(loff), "v"(goff), "s"(noise)
                 : "memory");
  };

  // Full-chunk DMA burst: compile-time trip count -> 32 back-to-back issues.
  auto issue_full = [&](int c) {
    const int buf = c & 1;
    unsigned goff = (unsigned)(c * TCHUNK) * gstride + (unsigned)bc * 8u;
#pragma unroll
    for (int kl = 0; kl < TCHUNK; ++kl) {
      issue_one(buf, kl, goff);
      goff += gstride;
    }
  };

  // Tail-chunk DMA burst (dynamic m < TCHUNK).
  auto issue_tail = [&](int c, int m) {
    const int buf = c & 1;
    unsigned goff = (unsigned)(c * TCHUNK) * gstride + (unsigned)bc * 8u;
    for (int kl = 0; kl < m; ++kl) {
      issue_one(buf, kl, goff);
      goff += gstride;
    }
  };

  // One Euler-Maruyama step; drifts from OLD state, phi uses old v.
  auto step = [&](int buf, int kl, float* outp) {
    const float2 z =
        *reinterpret_cast<const float2*>(&lds_noise[buf][kl][tid][0]);
    const float dv1 = i1 - beta1 * v1 - __sinf(phi1) + kap1 * (phi2 - phi1);
    const float dv2 = i2 - beta2 * v2 - __sinf(phi2) + kap2 * (phi1 - phi2);
    phi1 = __builtin_fmaf(v1, dt, phi1);
    v1   = __builtin_fmaf(dv1, dt, __builtin_fmaf(s1, z.x, v1));
    phi2 = __builtin_fmaf(v2, dt, phi2);
    v2   = __builtin_fmaf(dv2, dt, __builtin_fmaf(s2, z.y, v2));
    float4 yn; yn.x = phi1; yn.y = v1; yn.z = phi2; yn.w = v2;
    *reinterpret_cast<float4*>(outp) = yn;  // global_store_b128
  };

  // Prologue: DMA chunk 0.
  if (nchunks > 0) {
    if (full > 0) issue_full(0); else issue_tail(0, tail);
  }

  for (int c = 0; c < nchunks; ++c) {
    // Only chunk c's DMA is outstanding here.
    asm volatile("s_wait_asynccnt 0x0" ::: "memory");

    // Kick off chunk c+1's DMA so it overlaps chunk c's compute.
    const int nxt = c + 1;
    if (nxt < full)          issue_full(nxt);
    else if (nxt < nchunks)  issue_tail(nxt, tail);

    const int buf = c & 1;
    const int k0  = c * TCHUNK;

    if (active) {
      float* outp = out + ((size_t)b * N + (size_t)(k0 + 1)) * 4;
      if (c < full) {
        // Common path: compile-time 32 steps, unrolled to amortize control.
#pragma unroll 8
        for (int kl = 0; kl < TCHUNK; ++kl) {
          step(buf, kl, outp);
          outp += 4;
        }
      } else {
        for (int kl = 0; kl < tail; ++kl) {
          step(buf, kl, outp);
          outp += 4;
        }
      }
    }
  }
}

extern "C" void kernel_launch(void* const* d_in, const int* in_sizes, int n_in,
                              void* d_out, int out_size, void* d_ws, size_t ws_size,
                              hipStream_t stream) {
  const float* params = (const float*)d_in[0];
  const float* y0     = (const float*)d_in[1];
  const float* noise  = (const float*)d_in[2];
  const int*   Tp     = (const int*)d_in[3];
  const int*   Np     = (const int*)d_in[4];
  float*       out    = (float*)d_out;

  const int batch  = in_sizes[1] / 4;                 // y0 is [B, 4]
  const int blocks = (batch + TPB - 1) / TPB;

  jj_sde_kernel<<<blocks, TPB, 0, stream>>>(params, y0, noise, Tp, Np, out, batch);
}